// SparseChebBranch_89232240542461
// MI455X (gfx1250) — compile-verified
//
#include <hip/hip_runtime.h>
#include <hip/hip_bf16.h>
#include <stdint.h>

typedef __attribute__((ext_vector_type(2))) float v2f;
typedef __attribute__((ext_vector_type(8))) float v8f;

// ---------------------------------------------------------------------------
// Graph normalization: deg -> dinv -> w_norm
// ---------------------------------------------------------------------------
__global__ __launch_bounds__(256) void deg_kernel(float* __restrict__ deg,
                                                  const long long* __restrict__ src,
                                                  const long long* __restrict__ dst,
                                                  long long E) {
    long long e = (long long)blockIdx.x * blockDim.x + threadIdx.x;
    if (e >= E) return;
    long long s = src[e];
    if (s != dst[e]) atomicAdd(&deg[s], 1.0f);
}

__global__ __launch_bounds__(256) void dinv_kernel(const float* __restrict__ deg,
                                                   float* __restrict__ dinv, int n) {
    int i = blockIdx.x * blockDim.x + threadIdx.x;
    if (i >= n) return;
    float d = deg[i];
    dinv[i] = (d > 0.0f) ? rsqrtf(d) : 0.0f;
}

__global__ __launch_bounds__(256) void edgew_kernel(float* __restrict__ wnorm,
                                                    const float* __restrict__ dinv,
                                                    const long long* __restrict__ src,
                                                    const long long* __restrict__ dst,
                                                    long long E) {
    long long e = (long long)blockIdx.x * blockDim.x + threadIdx.x;
    if (e >= E) return;
    long long s = src[e], d = dst[e];
    // 2/lambda_max == 1 ; off-diagonal of scaled Laplacian is -dinv[s]*dinv[d]
    wnorm[e] = (s != d) ? (-dinv[s] * dinv[d]) : 0.0f;
}

// ---------------------------------------------------------------------------
// SpMM propagation: out[dst] += w * h[src]   (diag term == 0 for lambda_max=2)
// One wave32 per edge; lanes stride the feature dim; L2-resident f32 atomics.
// ---------------------------------------------------------------------------
__global__ __launch_bounds__(256) void spmm_kernel(float* __restrict__ out,
                                                   const float* __restrict__ h,
                                                   const long long* __restrict__ src,
                                                   const long long* __restrict__ dst,
                                                   const float* __restrict__ wnorm,
                                                   long long E, int d) {
    long long wave = (long long)blockIdx.x * (blockDim.x >> 5) + (threadIdx.x >> 5);
    int lane = threadIdx.x & 31;
    if (wave >= E) return;
    float w = wnorm[wave];
    if (w == 0.0f) return;
    const float* __restrict__ hs = h + (size_t)src[wave] * d;
    float* __restrict__ od = out + (size_t)dst[wave] * d;
    for (int i = lane; i < d; i += 32)
        atomicAdd(&od[i], w * hs[i]);
}

// Tx2 = 2*P - Tx0   (P updated in place)
__global__ __launch_bounds__(256) void tx2_kernel(float* __restrict__ P,
                                                  const float* __restrict__ Tx0,
                                                  size_t n) {
    size_t i = (size_t)blockIdx.x * blockDim.x + threadIdx.x;
    if (i >= n) return;
    P[i] = 2.0f * P[i] - Tx0[i];
}

// ---------------------------------------------------------------------------
// Fused Chebyshev GEMM:  out = Tx0@W[0] + Tx1@W[1] + Tx2@W[2] + bias
//
// Block = 8 waves = 128-row x 16-col output supertile (grid.y = N-tile).
// The [3][K][16] weight slice for this block's col0 is staged in LDS once,
// pair-interleaved: element (w, k, n) lives at ((w*K/2 + k/2)*16 + n)*2 + (k&1)
// so the B fragment for a k-step is two adjacent floats -> one ds_load_b64.
//
// Per wave: one 16x16 tile, K stepped by 4 with V_WMMA_F32_16X16X4_F32.
// A 16x4 layout (ISA 7.12.2): lanes 0-15 rows, VGPR pair K=0,1; lanes 16-31
//   K=2,3 -> per-lane float2 global load at row*K + 2*half + kk.
// B 4x16 layout: lanes stripe N, VGPR pair stripes K (half selects K=2,3).
// C/D 16x16: VGPR r -> row r (lanes 0-15) / row 8+r (lanes 16-31).
// ---------------------------------------------------------------------------
__global__ __launch_bounds__(256) void cheb_gemm_wmma(const float* __restrict__ A0,
                                                      const float* __restrict__ A1,
                                                      const float* __restrict__ A2,
                                                      const float* __restrict__ W,   // [3][K][Nd]
                                                      const float* __restrict__ bias,
                                                      float* __restrict__ out,
                                                      int M, int K, int Nd) {
    extern __shared__ float wlds[];   // 3*K*16 floats, pair-interleaved

    const int lane   = threadIdx.x & 31;
    const int wave   = threadIdx.x >> 5;
    const int tilesM = M >> 4;
    const int Khalf  = K >> 1;
    const int col0   = blockIdx.y << 4;

    // --- cooperative stage of W[:, :, col0:col0+16] into LDS ---
    for (int idx = threadIdx.x; idx < 3 * K * 16; idx += blockDim.x) {
        int w   = idx / (K * 16);
        int rem = idx - w * K * 16;
        int k   = rem >> 4;
        int n   = rem & 15;
        wlds[(((w * Khalf + (k >> 1)) << 4) + n) * 2 + (k & 1)] =
            W[((size_t)w * K + k) * Nd + col0 + n];
    }
    __syncthreads();

    const int tm = blockIdx.x * (blockDim.x >> 5) + wave;
    if (tm >= tilesM) return;
    const int row0 = tm << 4;

    const int half = lane >> 4;   // selects K sub-pair
    const int l16  = lane & 15;   // row (A) / col (B)

    v8f c = {};
#pragma unroll
    for (int w = 0; w < 3; ++w) {
        const float* __restrict__ A    = (w == 0) ? A0 : (w == 1) ? A1 : A2;
        const float* __restrict__ arow = A + (size_t)(row0 + l16) * K + (half << 1);
        const float* __restrict__ bbase = wlds + ((w * Khalf) << 5) + (l16 << 1)
                                               + (half << 5);
#pragma unroll 4
        for (int kk = 0; kk < K; kk += 4) {
            v2f a = *(const v2f*)(arow + kk);                      // global_load_b64
            v2f b = *(const v2f*)(bbase + ((kk >> 1) << 5));       // ds_load_b64
            c = __builtin_amdgcn_wmma_f32_16x16x4_f32(
                    /*neg_a=*/false, a, /*neg_b=*/false, b,
                    /*c_mod=*/(short)0, c, /*reuse_a=*/false, /*reuse_b=*/false);
        }
    }

    float bv = bias[col0 + l16];
#pragma unroll
    for (int r = 0; r < 8; ++r) {
        int row = row0 + r + (half << 3);
        out[(size_t)row * Nd + col0 + l16] = c[r] + bv;
    }
}

// ---------------------------------------------------------------------------
// In-place LayerNorm + ReLU, one wave32 per row, shuffle reductions.
// ---------------------------------------------------------------------------
__global__ __launch_bounds__(256) void layernorm_relu_kernel(float* __restrict__ h,
                                                             const float* __restrict__ g,
                                                             const float* __restrict__ bt,
                                                             int M, int d) {
    int row = blockIdx.x * (blockDim.x >> 5) + (threadIdx.x >> 5);
    if (row >= M) return;
    int lane = threadIdx.x & 31;
    float* __restrict__ hr = h + (size_t)row * d;

    float s = 0.0f;
    for (int i = lane; i < d; i += 32) s += hr[i];
#pragma unroll
    for (int off = 16; off > 0; off >>= 1) s += __shfl_xor(s, off, 32);
    float mu = s / (float)d;

    float v = 0.0f;
    for (int i = lane; i < d; i += 32) { float t = hr[i] - mu; v += t * t; }
#pragma unroll
    for (int off = 16; off > 0; off >>= 1) v += __shfl_xor(v, off, 32);
    float inv = rsqrtf(v / (float)d + 1e-5f);

    for (int i = lane; i < d; i += 32) {
        float t = (hr[i] - mu) * inv * g[i] + bt[i];
        hr[i] = (t > 0.0f) ? t : 0.0f;
    }
}

// ---------------------------------------------------------------------------
extern "C" void kernel_launch(void* const* d_in, const int* in_sizes, int n_in,
                              void* d_out, int out_size, void* d_ws, size_t ws_size,
                              hipStream_t stream) {
    const float*     x   = (const float*)d_in[0];
    const long long* ei  = (const long long*)d_in[1];   // int64 edge_index [2,E]
    const float*     W1  = (const float*)d_in[2];
    const float*     b1  = (const float*)d_in[3];
    const float*     g1  = (const float*)d_in[4];
    const float*     bt1 = (const float*)d_in[5];
    const float*     W2  = (const float*)d_in[6];
    const float*     b2  = (const float*)d_in[7];
    const float*     g2  = (const float*)d_in[8];
    const float*     bt2 = (const float*)d_in[9];
    float* out = (float*)d_out;

    const int D1 = 128, D2 = 256;
    const int N = in_sizes[0] / D1;                   // 50000
    const long long E = (long long)in_sizes[1] / 2;   // 800000
    const long long* src = ei;
    const long long* dst = ei + E;

    // Workspace carve (256B aligned slices)
    char* p = (char*)d_ws;
    auto take = [&](size_t bytes) -> float* {
        float* r = (float*)p;
        p += (bytes + 255) & ~(size_t)255;
        return r;
    };
    float* deg   = take((size_t)N * 4);
    float* dinv  = take((size_t)N * 4);
    float* wnorm = take((size_t)E * 4);
    float* bufA  = take((size_t)N * D2 * 4);   // Tx1
    float* bufB  = take((size_t)N * D2 * 4);   // prop(Tx1) -> Tx2
    float* bufC  = take((size_t)N * D2 * 4);   // layer1 out / h1 (LN in place)

    const int T = 256;
    dim3 blk(T);
    long long eBlocks = (E + T - 1) / T;
    long long wBlocks = (E + (T >> 5) - 1) / (T >> 5);   // one wave per edge
    const int tilesM    = N >> 4;                         // 3125
    const int mBlocks   = (tilesM + (T >> 5) - 1) / (T >> 5);

    // --- edge normalization ---
    hipMemsetAsync(deg, 0, (size_t)N * 4, stream);
    deg_kernel<<<(unsigned)eBlocks, blk, 0, stream>>>(deg, src, dst, E);
    dinv_kernel<<<(N + T - 1) / T, blk, 0, stream>>>(deg, dinv, N);
    edgew_kernel<<<(unsigned)eBlocks, blk, 0, stream>>>(wnorm, dinv, src, dst, E);

    // --- layer 1: 128 -> 256 ---
    hipMemsetAsync(bufA, 0, (size_t)N * D1 * 4, stream);
    spmm_kernel<<<(unsigned)wBlocks, blk, 0, stream>>>(bufA, x, src, dst, wnorm, E, D1);   // Tx1
    hipMemsetAsync(bufB, 0, (size_t)N * D1 * 4, stream);
    spmm_kernel<<<(unsigned)wBlocks, blk, 0, stream>>>(bufB, bufA, src, dst, wnorm, E, D1); // P
    {
        size_t n = (size_t)N * D1;
        tx2_kernel<<<(unsigned)((n + T - 1) / T), blk, 0, stream>>>(bufB, x, n);            // Tx2
    }
    {
        dim3 grid(mBlocks, D2 >> 4);
        size_t shmem = (size_t)3 * D1 * 16 * 4;   // 24 KB
        cheb_gemm_wmma<<<grid, blk, shmem, stream>>>(x, bufA, bufB, W1, b1, bufC, N, D1, D2);
    }
    layernorm_relu_kernel<<<(N + (T >> 5) - 1) / (T >> 5), blk, 0, stream>>>(bufC, g1, bt1, N, D2);

    // --- layer 2: 256 -> 128 ---
    hipMemsetAsync(bufA, 0, (size_t)N * D2 * 4, stream);
    spmm_kernel<<<(unsigned)wBlocks, blk, 0, stream>>>(bufA, bufC, src, dst, wnorm, E, D2);  // Tx1
    hipMemsetAsync(bufB, 0, (size_t)N * D2 * 4, stream);
    spmm_kernel<<<(unsigned)wBlocks, blk, 0, stream>>>(bufB, bufA, src, dst, wnorm, E, D2);  // P
    {
        size_t n = (size_t)N * D2;
        tx2_kernel<<<(unsigned)((n + T - 1) / T), blk, 0, stream>>>(bufB, bufC, n);          // Tx2
    }
    {
        dim3 grid(mBlocks, D1 >> 4);
        size_t shmem = (size_t)3 * D2 * 16 * 4;   // 48 KB
        cheb_gemm_wmma<<<grid, blk, shmem, stream>>>(bufC, bufA, bufB, W2, b2, out, N, D2, D1);
    }
    layernorm_relu_kernel<<<(N + (T >> 5) - 1) / (T >> 5), blk, 0, stream>>>(out, g2, bt2, N, D1);
}